// DySAT_33586644255251
// MI455X (gfx1250) — compile-verified
//
#include <hip/hip_runtime.h>

// ---------------- problem constants (match reference) ----------------
#define TT   16        // snapshots
#define NN   50000     // nodes per snapshot
#define EE   800000    // edges per snapshot
#define FIN  128       // input features
#define HH   4         // heads
#define DD   4         // per-head dim  (H*D = 16 output features)
#define HD   16        // H*D

typedef __attribute__((ext_vector_type(2))) float v2f;
typedef __attribute__((ext_vector_type(8))) float v8f;

// =====================================================================
// Kernel 1: h = x @ W + b  via V_WMMA_F32_16X16X4_F32, and out = h
// One wave (32 lanes) computes one 16-row x 16-col tile, K=128 in 32 steps.
// Rows are the flattened (t, n) index: 800000 rows = 50000 tiles exactly.
// =====================================================================
__global__ __launch_bounds__(256)
void dysat_gemm_wmma(const float* __restrict__ x,     // [T*N, 128]
                     const float* __restrict__ W,     // [128, 16]
                     const float* __restrict__ bias,  // [16]
                     float* __restrict__ h,           // [T*N, 16]
                     float* __restrict__ out)         // [T*N, 16] (residual init)
{
    const int wave = blockIdx.x * (blockDim.x >> 5) + (threadIdx.x >> 5);
    const int lane = threadIdx.x & 31;
    const int half = lane >> 4;      // 0: lanes 0-15, 1: lanes 16-31
    const int l    = lane & 15;

    const long row0 = (long)wave * 16;
    const float* __restrict__ xA = x + (row0 + l) * FIN + 2 * half;

    v8f acc = {0.f, 0.f, 0.f, 0.f, 0.f, 0.f, 0.f, 0.f};

#pragma unroll
    for (int k = 0; k < FIN / 4; ++k) {
        // A (16x4 f32): lane holds row M=l, K = 4k + 2*half + {0,1}
        const float2 a2 = *(const float2*)(xA + 4 * k);
        v2f a; a.x = a2.x; a.y = a2.y;
        // B (4x16 f32): mirrored layout, lane holds col N=l,
        // K rows 4k + 2*half + {0,1};  W row stride = 16 floats
        const int kr = 4 * k + 2 * half;
        v2f bm;
        bm.x = W[kr * HD + l];
        bm.y = W[(kr + 1) * HD + l];
        acc = __builtin_amdgcn_wmma_f32_16x16x4_f32(
            /*neg_a=*/false, a, /*neg_b=*/false, bm,
            /*c_mod=*/(short)0, acc, /*reuse_a=*/false, /*reuse_b=*/false);
    }

    // C/D layout: VGPR v -> row M = v + 8*half, col N = l
    const float bc = bias[l];
#pragma unroll
    for (int v = 0; v < 8; ++v) {
        const long row = row0 + v + 8 * half;
        const float val = acc[v] + bc;
        h[row * HD + l]   = val;
        out[row * HD + l] = val;   // residual: out starts as h
    }
}

// =====================================================================
// Kernel 2: per (t,n,head) attention logits + denom zero-init
// alpha_l[tn,hh] = dot(h[tn, hh*4 .. +3], a_l[hh,:]) ; same for alpha_r
// =====================================================================
__global__ __launch_bounds__(256)
void dysat_alpha(const float* __restrict__ h,     // [T*N, 16]
                 const float* __restrict__ a_l,   // [4,4]
                 const float* __restrict__ a_r,   // [4,4]
                 float* __restrict__ alpha_l,     // [T*N, 4]
                 float* __restrict__ alpha_r,     // [T*N, 4]
                 float* __restrict__ denom)       // [T*N, 4]
{
    const int idx = blockIdx.x * blockDim.x + threadIdx.x;   // over T*N*H
    if (idx >= TT * NN * HH) return;
    const int hh   = idx & 3;
    const long tn  = idx >> 2;

    const float4 hv = *(const float4*)(h + tn * HD + hh * DD);
    const float4 al = *(const float4*)(a_l + hh * DD);
    const float4 ar = *(const float4*)(a_r + hh * DD);

    alpha_l[idx] = hv.x * al.x + hv.y * al.y + hv.z * al.z + hv.w * al.w;
    alpha_r[idx] = hv.x * ar.x + hv.y * ar.y + hv.z * ar.z + hv.w * ar.w;
    denom[idx]   = 0.0f;
}

__device__ __forceinline__ float lrelu02(float s) {
    return s > 0.0f ? s : 0.2f * s;
}

// =====================================================================
// Kernel 3: edge pass 1 -- accumulate softmax denominators
// e = lrelu(alpha_l[dst] + alpha_r[src]); w = exp(e - max_over_heads(e))
// denom[dst] += w   (HW f32 atomics; denom is L2-resident, 12.8 MB)
// =====================================================================
__global__ __launch_bounds__(256)
void dysat_edge_denom(const int* __restrict__ eidx,      // [T, 2, E] int32
                      const float* __restrict__ alpha_l, // [T*N, 4]
                      const float* __restrict__ alpha_r, // [T*N, 4]
                      float* __restrict__ denom)         // [T*N, 4]
{
    const long tid = (long)blockIdx.x * blockDim.x + threadIdx.x;  // over T*E
    if (tid >= (long)TT * EE) return;
    const int t = (int)(tid / EE);
    const int e = (int)(tid - (long)t * EE);
    const int* __restrict__ base = eidx + (long)t * 2 * EE;
    const long nd = (long)t * NN + base[e];        // dst
    const long ns = (long)t * NN + base[EE + e];   // src

    const float4 al = *(const float4*)(alpha_l + nd * 4);
    const float4 ar = *(const float4*)(alpha_r + ns * 4);

    const float s0 = lrelu02(al.x + ar.x);
    const float s1 = lrelu02(al.y + ar.y);
    const float s2 = lrelu02(al.z + ar.z);
    const float s3 = lrelu02(al.w + ar.w);
    const float m  = fmaxf(fmaxf(s0, s1), fmaxf(s2, s3));

    float* d = denom + nd * 4;
    unsafeAtomicAdd(d + 0, __expf(s0 - m));
    unsafeAtomicAdd(d + 1, __expf(s1 - m));
    unsafeAtomicAdd(d + 2, __expf(s2 - m));
    unsafeAtomicAdd(d + 3, __expf(s3 - m));
}

// =====================================================================
// Kernel 4: edge pass 2 -- weighted aggregation
// alpha = w / denom[dst];  out[dst] += h[src] * alpha (per head)
// =====================================================================
__global__ __launch_bounds__(256)
void dysat_edge_aggr(const int* __restrict__ eidx,
                     const float* __restrict__ alpha_l,
                     const float* __restrict__ alpha_r,
                     const float* __restrict__ denom,
                     const float* __restrict__ h,     // [T*N, 16]
                     float* __restrict__ out)         // [T*N, 16]
{
    const long tid = (long)blockIdx.x * blockDim.x + threadIdx.x;  // over T*E
    if (tid >= (long)TT * EE) return;
    const int t = (int)(tid / EE);
    const int e = (int)(tid - (long)t * EE);
    const int* __restrict__ base = eidx + (long)t * 2 * EE;
    const long nd = (long)t * NN + base[e];        // dst
    const long ns = (long)t * NN + base[EE + e];   // src

    const float4 al = *(const float4*)(alpha_l + nd * 4);
    const float4 ar = *(const float4*)(alpha_r + ns * 4);
    const float4 dn = *(const float4*)(denom   + nd * 4);

    const float s0 = lrelu02(al.x + ar.x);
    const float s1 = lrelu02(al.y + ar.y);
    const float s2 = lrelu02(al.z + ar.z);
    const float s3 = lrelu02(al.w + ar.w);
    const float m  = fmaxf(fmaxf(s0, s1), fmaxf(s2, s3));

    float wgt[HH];
    wgt[0] = __expf(s0 - m) / dn.x;
    wgt[1] = __expf(s1 - m) / dn.y;
    wgt[2] = __expf(s2 - m) / dn.z;
    wgt[3] = __expf(s3 - m) / dn.w;

    const float* __restrict__ hs = h + ns * HD;   // gather: L2-resident
    float* __restrict__ od = out + nd * HD;
#pragma unroll
    for (int hh = 0; hh < HH; ++hh) {
        const float4 hv = *(const float4*)(hs + hh * DD);
        const float a = wgt[hh];
        unsafeAtomicAdd(od + hh * DD + 0, hv.x * a);
        unsafeAtomicAdd(od + hh * DD + 1, hv.y * a);
        unsafeAtomicAdd(od + hh * DD + 2, hv.z * a);
        unsafeAtomicAdd(od + hh * DD + 3, hv.w * a);
    }
}

// =====================================================================
// Host-side launcher
// =====================================================================
extern "C" void kernel_launch(void* const* d_in, const int* in_sizes, int n_in,
                              void* d_out, int out_size, void* d_ws, size_t ws_size,
                              hipStream_t stream) {
    const float* x    = (const float*)d_in[0];   // [T, N, 128]
    const int*   eidx = (const int*)  d_in[1];   // [T, 2, E] (JAX x64 off -> int32)
    const float* W    = (const float*)d_in[2];   // [128, 16]
    const float* bias = (const float*)d_in[3];   // [16]
    const float* a_l  = (const float*)d_in[4];   // [4, 4]
    const float* a_r  = (const float*)d_in[5];   // [4, 4]
    float* out = (float*)d_out;                  // [T, N, 16]

    // workspace layout (floats)
    float* ws      = (float*)d_ws;
    float* h       = ws;                                  // T*N*16 = 12.8M
    float* alpha_l = h       + (long)TT * NN * HD;        // T*N*4
    float* alpha_r = alpha_l + (long)TT * NN * HH;        // T*N*4
    float* denom   = alpha_r + (long)TT * NN * HH;        // T*N*4

    // 1) GEMM (WMMA): 800000 rows / 16-row tile = 50000 waves; 8 waves/block
    {
        const int tiles = (TT * NN) / 16;        // 50000
        const int wavesPerBlock = 8;             // 256 threads
        dysat_gemm_wmma<<<tiles / wavesPerBlock, 256, 0, stream>>>(
            x, W, bias, h, out);
    }
    // 2) attention logits + denom init: T*N*H threads
    {
        const long total = (long)TT * NN * HH;
        dysat_alpha<<<(int)((total + 255) / 256), 256, 0, stream>>>(
            h, a_l, a_r, alpha_l, alpha_r, denom);
    }
    // 3) edge pass 1: T*E threads
    {
        const long total = (long)TT * EE;
        dysat_edge_denom<<<(int)((total + 255) / 256), 256, 0, stream>>>(
            eidx, alpha_l, alpha_r, denom);
    }
    // 4) edge pass 2: T*E threads
    {
        const long total = (long)TT * EE;
        dysat_edge_aggr<<<(int)((total + 255) / 256), 256, 0, stream>>>(
            eidx, alpha_l, alpha_r, denom, h, out);
    }
}